// LightGCN_47132971106661
// MI455X (gfx1250) — compile-verified
//
#include <hip/hip_runtime.h>

#define N_NODES 100000
#define N_EDGES 1250000
#define DIM 64

typedef __attribute__((ext_vector_type(2))) float v2f;
typedef __attribute__((ext_vector_type(8))) float v8f;

// ---------------------------------------------------------------- utilities

__global__ void zero_f32_kernel(float* __restrict__ p, int n) {
    int i = blockIdx.x * blockDim.x + threadIdx.x;
    if (i < n) p[i] = 0.0f;
}

// deg[i] = in-degree over dst
__global__ void degree_kernel(const int* __restrict__ dst,
                              float* __restrict__ deg) {
    int e = blockIdx.x * blockDim.x + threadIdx.x;
    if (e < N_EDGES) atomicAdd(&deg[dst[e]], 1.0f);
}

// dinv[i] = deg>0 ? rsqrt(deg) : 0
__global__ void dinv_kernel(const float* __restrict__ deg,
                            float* __restrict__ dinv) {
    int i = blockIdx.x * blockDim.x + threadIdx.x;
    if (i < N_NODES) {
        float d = deg[i];
        dinv[i] = (d > 0.0f) ? rsqrtf(d) : 0.0f;
    }
}

// norm[e] = dinv[src[e]] * dinv[dst[e]]   (computed once, reused 3 layers)
__global__ void norm_kernel(const int* __restrict__ src,
                            const int* __restrict__ dst,
                            const float* __restrict__ dinv,
                            float* __restrict__ norm) {
    int e = blockIdx.x * blockDim.x + threadIdx.x;
    if (e < N_EDGES) norm[e] = dinv[src[e]] * dinv[dst[e]];
}

// out[i] = alpha[0] * emb[i]
__global__ void init_out_kernel(const float* __restrict__ emb,
                                const float* __restrict__ alpha,
                                float* __restrict__ out) {
    int i = blockIdx.x * blockDim.x + threadIdx.x;
    if (i < N_NODES * DIM) out[i] = alpha[0] * emb[i];
}

// One propagation layer: x_next[dst] += x_in[src] * norm_e
// 16 threads per edge, float4 per thread -> fully coalesced 256B row gather,
// 4x global_atomic_add_f32 scatter per thread.
__global__ void scatter_kernel(const float* __restrict__ xin,
                               const int* __restrict__ src,
                               const int* __restrict__ dst,
                               const float* __restrict__ norm,
                               float* __restrict__ xout) {
    int t = blockIdx.x * blockDim.x + threadIdx.x;
    int e = t >> 4;          // edge
    int c = t & 15;          // float4 chunk within the 64-float row
    if (e >= N_EDGES) return;
    int s = src[e];
    int d = dst[e];
    float nrm = norm[e];
    const float4 v = *(const float4*)(xin + (long)s * DIM + c * 4);
    float* o = xout + (long)d * DIM + c * 4;
    atomicAdd(o + 0, v.x * nrm);
    atomicAdd(o + 1, v.y * nrm);
    atomicAdd(o + 2, v.z * nrm);
    atomicAdd(o + 3, v.w * nrm);
}

// out[i] += alpha[layer] * x[i]
__global__ void accum_kernel(const float* __restrict__ x,
                             const float* __restrict__ alpha, int li,
                             float* __restrict__ out) {
    int i = blockIdx.x * blockDim.x + threadIdx.x;
    if (i < N_NODES * DIM) out[i] += alpha[li] * x[i];
}

// ---------------------------------------------------------------- final dots
//
// Per-edge score = out[src] . out[dst], 64-d f32. One wave handles a tile of
// 16 edges via exact-f32 WMMA: A = out[src rows] (16x64), B = out[dst cols]
// (64x16), accumulate D += A_k x B_k over 16 chunks of K=4 with
// v_wmma_f32_16x16x4_f32; the 16 scores are diag(D).
//
// 32-bit A 16x4 fragment layout (ISA 7.12.2): lane l holds row M = l&15,
// VGPR0/1 = K {0,1} (lanes 0-15) or K {2,3} (lanes 16-31)  ->  each lane
// loads float2 at element 4k + 2*(l>>4) of its row. B 4x16 is symmetric.
__global__ void edge_dot_wmma_kernel(const float* __restrict__ emb,
                                     const int* __restrict__ src,
                                     const int* __restrict__ dst,
                                     float* __restrict__ out) {
    const int lane = threadIdx.x & 31;
    const int wave = (blockIdx.x * blockDim.x + threadIdx.x) >> 5;
    const int NTILES = N_EDGES / 16;      // 78125, exact
    if (wave >= NTILES) return;           // wave-uniform: EXEC all-ones below

    const int sub = lane & 15;            // edge within tile (= M = N)
    const int hi  = lane >> 4;            // K half select
    const int e   = wave * 16 + sub;
    const long srow = (long)src[e] * DIM;
    const long drow = (long)dst[e] * DIM;
    const int off0 = 2 * hi;

    v8f c = {};
#pragma unroll
    for (int k = 0; k < 16; ++k) {
        v2f a = *(const v2f*)(emb + srow + 4 * k + off0);   // global_load_b64
        v2f b = *(const v2f*)(emb + drow + 4 * k + off0);   // global_load_b64
        c = __builtin_amdgcn_wmma_f32_16x16x4_f32(
                /*neg_a=*/false, a, /*neg_b=*/false, b,
                /*c_mod=*/(short)0, c, /*reuse_a=*/false, /*reuse_b=*/false);
    }

    // diag(D): e<8 -> VGPR e @ lane e ; e>=8 -> VGPR e-8 @ lane e+16
    float* obase = out + wave * 16;
#pragma unroll
    for (int r = 0; r < 8; ++r) {
        if (lane == r)      obase[r]     = c[r];
        if (lane == r + 24) obase[r + 8] = c[r];
    }
}

// ---------------------------------------------------------------- launcher

extern "C" void kernel_launch(void* const* d_in, const int* in_sizes, int n_in,
                              void* d_out, int out_size, void* d_ws, size_t ws_size,
                              hipStream_t stream) {
    const float* emb   = (const float*)d_in[0];   // [N_NODES, 64]
    const float* alpha = (const float*)d_in[1];   // [4]
    const int*   eidx  = (const int*)d_in[2];     // [2, N_EDGES]
    const int*   src   = eidx;
    const int*   dst   = eidx + N_EDGES;
    float*       scores = (float*)d_out;          // [N_EDGES]

    // workspace layout (floats)
    float* ws    = (float*)d_ws;
    float* deg   = ws;                              // N_NODES
    float* dinv  = ws + N_NODES;                    // N_NODES
    float* norm  = ws + 2 * N_NODES;                // N_EDGES
    float* xA    = norm + N_EDGES;                  // N_NODES*64
    float* xB    = xA + (long)N_NODES * DIM;        // N_NODES*64
    float* oacc  = xB + (long)N_NODES * DIM;        // N_NODES*64

    const int BLK = 256;
    const int nND = (N_NODES + BLK - 1) / BLK;                 // node grid
    const int nNE = (N_EDGES + BLK - 1) / BLK;                 // edge grid
    const int nEL = (N_NODES * DIM + BLK - 1) / BLK;           // element grid
    const int nSC = (N_EDGES * 16 + BLK - 1) / BLK;            // scatter grid

    // degrees -> dinv -> per-edge norm
    zero_f32_kernel<<<nND, BLK, 0, stream>>>(deg, N_NODES);
    degree_kernel<<<nNE, BLK, 0, stream>>>(dst, deg);
    dinv_kernel<<<nND, BLK, 0, stream>>>(deg, dinv);
    norm_kernel<<<nNE, BLK, 0, stream>>>(src, dst, dinv, norm);

    // out = alpha0 * emb
    init_out_kernel<<<nEL, BLK, 0, stream>>>(emb, alpha, oacc);

    // 3 propagation layers (ping-pong xA/xB; layer 0 reads emb directly)
    const float* xin = emb;
    float* bufs[2] = { xA, xB };
    for (int l = 0; l < 3; ++l) {
        float* xnext = bufs[l & 1];
        zero_f32_kernel<<<nEL, BLK, 0, stream>>>(xnext, N_NODES * DIM);
        scatter_kernel<<<nSC, BLK, 0, stream>>>(xin, src, dst, norm, xnext);
        accum_kernel<<<nEL, BLK, 0, stream>>>(xnext, alpha, l + 1, oacc);
        xin = xnext;
    }

    // per-edge dots via exact-f32 WMMA, 16 edges per wave
    const int waves  = N_EDGES / 16;          // 78125
    const int wavesPerBlk = BLK / 32;         // 8
    const int nDOT = (waves + wavesPerBlk - 1) / wavesPerBlk;
    edge_dot_wmma_kernel<<<nDOT, BLK, 0, stream>>>(oacc, src, dst, scores);
}